// DDA_PU_loss_52029233824478
// MI455X (gfx1250) — compile-verified
//
#include <hip/hip_runtime.h>
#include <hip/hip_bf16.h>

typedef __attribute__((ext_vector_type(2))) float v2f;
typedef __attribute__((ext_vector_type(8))) float v8f;
typedef __attribute__((ext_vector_type(4))) int   v4i;

#define NB   1024      // stage-1 blocks (multiple of 64 for the WMMA reduce)
#define NT   256       // 8 wave32s per block
#define NVIR 16384     // row stride of the dense matrices (per reference)

// ---------------------------------------------------------------------------
// Stage 1: gather squared residuals at (pos, neg) coordinates, produce one
// deterministic partial-sum pair per block.
//   - 1024 x 256 wave32 threads = 262k concurrent gather streams (latency
//     hiding is the whole game at 23.3 TB/s with a 1 GB random footprint).
//   - Index streams read 4 pairs at a time via 128-bit non-temporal loads:
//     4x fewer index instructions, 8 independent gathers in flight per
//     iteration, and no L2 pollution (matrix lines keep their 192 MB L2
//     residency; ~20% of the two matrices fits).
//   - 32-bit address math: flat index < 2^27, so no 64-bit mads.
// ---------------------------------------------------------------------------
__global__ __launch_bounds__(NT) void dda_gather_partials(
    const float* __restrict__ recon, const float* __restrict__ dv,
    const int* __restrict__ px, const int* __restrict__ py,
    const int* __restrict__ nx, const int* __restrict__ ny,
    int npos, int nneg,
    float* __restrict__ ws_pos, float* __restrict__ ws_neg)
{
    const int stride = gridDim.x * blockDim.x;
    const int gid    = blockIdx.x * blockDim.x + threadIdx.x;

    float pacc = 0.0f;
    {
        const int nvec = npos >> 2;
        const v4i* __restrict__ px4 = (const v4i*)px;
        const v4i* __restrict__ py4 = (const v4i*)py;
        for (int i = gid; i < nvec; i += stride) {
            const v4i xs = __builtin_nontemporal_load(px4 + i);
            const v4i ys = __builtin_nontemporal_load(py4 + i);
            #pragma unroll
            for (int k = 0; k < 4; ++k) {
                const int a = xs[k] * NVIR + ys[k];
                const float d = recon[a] - dv[a];
                pacc = fmaf(d, d, pacc);
            }
        }
        for (int i = (nvec << 2) + gid; i < npos; i += stride) {
            const int a = px[i] * NVIR + py[i];
            const float d = recon[a] - dv[a];
            pacc = fmaf(d, d, pacc);
        }
    }

    float nacc = 0.0f;
    {
        const int nvec = nneg >> 2;
        const v4i* __restrict__ nx4 = (const v4i*)nx;
        const v4i* __restrict__ ny4 = (const v4i*)ny;
        for (int i = gid; i < nvec; i += stride) {
            const v4i xs = __builtin_nontemporal_load(nx4 + i);
            const v4i ys = __builtin_nontemporal_load(ny4 + i);
            #pragma unroll
            for (int k = 0; k < 4; ++k) {
                const int a = xs[k] * NVIR + ys[k];
                const float d = recon[a] - dv[a];
                nacc = fmaf(d, d, nacc);
            }
        }
        for (int i = (nvec << 2) + gid; i < nneg; i += stride) {
            const int a = nx[i] * NVIR + ny[i];
            const float d = recon[a] - dv[a];
            nacc = fmaf(d, d, nacc);
        }
    }

    // wave32 butterfly reduce (deterministic tree)
    for (int m = 16; m >= 1; m >>= 1) {
        pacc += __shfl_xor(pacc, m, 32);
        nacc += __shfl_xor(nacc, m, 32);
    }

    __shared__ float sp[NT / 32];
    __shared__ float sn[NT / 32];
    const int lane = threadIdx.x & 31;
    const int wave = threadIdx.x >> 5;
    if (lane == 0) { sp[wave] = pacc; sn[wave] = nacc; }
    __syncthreads();

    if (threadIdx.x == 0) {
        float tp = 0.0f, tn = 0.0f;
        for (int w = 0; w < NT / 32; ++w) { tp += sp[w]; tn += sn[w]; }
        ws_pos[blockIdx.x] = tp;
        ws_neg[blockIdx.x] = tn;
    }
}

// ---------------------------------------------------------------------------
// Stage 2: one wave32. Reduce the NB per-block partials with the f32 matrix
// pipe: D = ones(16x4) x B(4x16) + C accumulates column sums in full f32.
// With an all-ones A, every slot of B contributes exactly once, so the sum
// of all 256 D entries equals 16x the grand total independent of the exact
// B-operand VGPR striping.
// ---------------------------------------------------------------------------
__global__ __launch_bounds__(32) void dda_finalize_wmma(
    const float* __restrict__ ws_pos, const float* __restrict__ ws_neg,
    const float* __restrict__ alpha_p, float* __restrict__ out, int nb)
{
    const int lane = threadIdx.x;

    v2f a; a[0] = 1.0f; a[1] = 1.0f;          // A = ones(16x4)
    v8f cpos = {};                            // f32 accumulators
    v8f cneg = {};

    for (int base = 0; base < nb; base += 64) {
        v2f bp, bn;                           // 64 partials -> B(4x16)
        bp[0] = ws_pos[base + lane];
        bp[1] = ws_pos[base + 32 + lane];
        bn[0] = ws_neg[base + lane];
        bn[1] = ws_neg[base + 32 + lane];
        cpos = __builtin_amdgcn_wmma_f32_16x16x4_f32(
            false, a, false, bp, (short)0, cpos, false, false);
        cneg = __builtin_amdgcn_wmma_f32_16x16x4_f32(
            false, a, false, bn, (short)0, cneg, false, false);
    }

    // Sum the whole 16x16 D across the wave: 8 VGPRs x 32 lanes = 16 * total.
    float ps = 0.0f, ns = 0.0f;
    #pragma unroll
    for (int i = 0; i < 8; ++i) { ps += cpos[i]; ns += cneg[i]; }
    for (int m = 16; m >= 1; m >>= 1) {
        ps += __shfl_xor(ps, m, 32);
        ns += __shfl_xor(ns, m, 32);
    }
    ps *= (1.0f / 16.0f);
    ns *= (1.0f / 16.0f);

    if (lane == 0) {
        const float alpha = alpha_p[0];
        out[0] = ps * ((1.0f - alpha) * 0.5f) + ns * (alpha * 0.5f);
    }
}

extern "C" void kernel_launch(void* const* d_in, const int* in_sizes, int n_in,
                              void* d_out, int out_size, void* d_ws, size_t ws_size,
                              hipStream_t stream) {
    const float* recon = (const float*)d_in[0];   // drug_virus_reconstruct
    const float* dv    = (const float*)d_in[1];   // drug_virus
    // d_in[2] = drug_virus_mask (unused by the reference math)
    const int* px = (const int*)d_in[3];
    const int* py = (const int*)d_in[4];
    const int* nx = (const int*)d_in[5];
    const int* ny = (const int*)d_in[6];
    const float* alpha = (const float*)d_in[7];

    const int npos = in_sizes[3];
    const int nneg = in_sizes[5];

    float* ws_pos = (float*)d_ws;      // NB floats
    float* ws_neg = ws_pos + NB;       // NB floats (8 KB total)

    dda_gather_partials<<<NB, NT, 0, stream>>>(recon, dv, px, py, nx, ny,
                                               npos, nneg, ws_pos, ws_neg);
    dda_finalize_wmma<<<1, 32, 0, stream>>>(ws_pos, ws_neg, alpha,
                                            (float*)d_out, NB);
}